// _CoxSquareLoss_15187004358711
// MI455X (gfx1250) — compile-verified
//
#include <hip/hip_runtime.h>
#include <math.h>

// Problem constants (match reference)
#define N_ELEM   8388608
#define T_MAX    1000000
#define SCAN_BLK 1024
#define NB       ((T_MAX + SCAN_BLK - 1) / SCAN_BLK)   // 977 chunks

typedef float v4f __attribute__((ext_vector_type(4)));

// Workspace layout (bytes):
//   A      float [T_MAX]  @ 0          (4,000,000)
//   E      float [T_MAX]  @ 4,000,000  (4,000,000)
//   acc    double[2]      @ 8,000,000  (16)   acc[0]=sum ev*x, acc[1]=sum E*logS
//   Bsum   double[1024]   @ 8,000,016
//   Boff   double[1024]   @ 8,008,208
// total 8,016,400 bytes
#define ZERO_UINT4 500001   // (A + E + acc) = 8,000,016 B = 500,001 * 16

__device__ __forceinline__ void atomic_add_f32(float* p, float v) {
#if defined(__HIP_DEVICE_COMPILE__)
    unsafeAtomicAdd(p, v);          // -> global_atomic_add_f32
#else
    atomicAdd(p, v);
#endif
}
__device__ __forceinline__ void atomic_add_f64(double* p, double v) {
#if defined(__HIP_DEVICE_COMPILE__)
    unsafeAtomicAdd(p, v);          // -> global_atomic_add_f64
#else
    atomicAdd(p, v);
#endif
}

// ---------------------------------------------------------------- zeroing
__global__ void cox_zero(v4f* __restrict__ ws) {
    int i = blockIdx.x * blockDim.x + threadIdx.x;
    if (i < ZERO_UINT4) {
        v4f z = { 0.f, 0.f, 0.f, 0.f };
        ws[i] = z;
    }
}

// ---------------------------------------------------------------- histogram
// x4: N/4 float4 of x.  tg4: N/2 float4 of target, layout (t0,e0,t1,e1).
__global__ void cox_hist(const v4f* __restrict__ x4,
                         const v4f* __restrict__ tg4,
                         float* __restrict__ A,
                         float* __restrict__ E,
                         double* __restrict__ acc,
                         int n4) {
    int gid = blockIdx.x * blockDim.x + threadIdx.x;
    float sum_evx = 0.0f;
    if (gid < n4) {
        // stream with NT hint: read-once data must not evict the L2-resident
        // A/E histograms that the atomics hammer
        v4f xv = __builtin_nontemporal_load(&x4[gid]);
        v4f ta = __builtin_nontemporal_load(&tg4[2 * gid]);
        v4f tb = __builtin_nontemporal_load(&tg4[2 * gid + 1]);
        float xs[4] = { xv.x, xv.y, xv.z, xv.w };
        float ts[4] = { ta.x, ta.z, tb.x, tb.z };
        float es[4] = { ta.y, ta.w, tb.y, tb.w };
#pragma unroll
        for (int k = 0; k < 4; ++k) {
            int t = (int)ts[k];
            atomic_add_f32(&A[t], __expf(xs[k]));
            if (es[k] != 0.0f) {
                atomic_add_f32(&E[t], 1.0f);
                sum_evx += xs[k];            // ev is exactly 0 or 1
            }
        }
    }
    // wave32 reduction
#pragma unroll
    for (int o = 16; o > 0; o >>= 1)
        sum_evx += __shfl_xor(sum_evx, o, 32);
    __shared__ float wsum[8];
    int lane = threadIdx.x & 31;
    int wv   = threadIdx.x >> 5;
    if (lane == 0) wsum[wv] = sum_evx;
    __syncthreads();
    if (threadIdx.x == 0) {
        float s = 0.0f;
#pragma unroll
        for (int i = 0; i < 8; ++i) s += wsum[i];
        atomic_add_f64(&acc[0], (double)s);
    }
}

// ---------------------------------------------------------------- chunk sums
__global__ void cox_block_sum(const float* __restrict__ A,
                              double* __restrict__ Bsum) {
    __shared__ double s[256];
    int b    = blockIdx.x;
    int base = b * SCAN_BLK;
    double v = 0.0;
    for (int i = threadIdx.x; i < SCAN_BLK; i += 256) {
        int t = base + i;
        if (t < T_MAX) v += (double)A[t];
    }
    s[threadIdx.x] = v;
    __syncthreads();
    for (int st = 128; st > 0; st >>= 1) {
        if (threadIdx.x < st) s[threadIdx.x] += s[threadIdx.x + st];
        __syncthreads();
    }
    if (threadIdx.x == 0) Bsum[b] = s[0];
}

// ---------------------------------------------------------------- inter-chunk suffix scan (1 block)
__global__ void cox_scan_blocks(const double* __restrict__ Bsum,
                                double* __restrict__ Boff) {
    __shared__ double s[1024];
    int i = threadIdx.x;
    double mine = (i < NB) ? Bsum[i] : 0.0;
    s[i] = mine;
    __syncthreads();
#pragma unroll
    for (int st = 1; st < 1024; st <<= 1) {
        double add = (i + st < 1024) ? s[i + st] : 0.0;
        __syncthreads();
        s[i] += add;
        __syncthreads();
    }
    if (i < NB) Boff[i] = s[i] - mine;   // exclusive: sum over chunks > i
}

// ---------------------------------------------------------------- intra-chunk suffix scan + E*logS
__global__ void cox_suffix_logsum(const float* __restrict__ A,
                                  const float* __restrict__ E,
                                  const double* __restrict__ Boff,
                                  double* __restrict__ acc) {
    __shared__ __attribute__((aligned(16))) float sA[SCAN_BLK];
    __shared__ double s[SCAN_BLK];
    int b = blockIdx.x;
    int i = threadIdx.x;
    int t = b * SCAN_BLK + i;

    // Stage this chunk of A into LDS through the CDNA5 async global->LDS
    // engine, 128b per request: lanes 0..255 each fetch 4 buckets.
    // the LDS address comes from the VDST VGPR; generic shared-ptr low 32 bits give the
    // workgroup-relative LDS byte offset (ISA 10.2 aperture mapping).
    // Last chunk has exactly 576 valid buckets (multiple of 4), so a simple
    // t0 < T_MAX guard is group-exact.
    if (i < SCAN_BLK / 4) {
        int t0 = b * SCAN_BLK + 4 * i;
        if (t0 < T_MAX) {
            unsigned lds_addr = (unsigned)(size_t)(&sA[4 * i]);
            unsigned long long gaddr = (unsigned long long)(size_t)(A + t0);
            asm volatile("global_load_async_to_lds_b128 %0, %1, off"
                         :: "v"(lds_addr), "v"(gaddr) : "memory");
        } else {
            sA[4 * i + 0] = 0.0f;   // disjoint LDS addrs: no race with async
            sA[4 * i + 1] = 0.0f;
            sA[4 * i + 2] = 0.0f;
            sA[4 * i + 3] = 0.0f;
        }
    }
    asm volatile("s_wait_asynccnt 0" ::: "memory");
    __syncthreads();

    double mine = (double)sA[i];
    s[i] = mine;
    __syncthreads();
#pragma unroll
    for (int st = 1; st < SCAN_BLK; st <<= 1) {
        double add = (i + st < SCAN_BLK) ? s[i + st] : 0.0;
        __syncthreads();
        s[i] += add;
        __syncthreads();
    }
    // S(t) = suffix within chunk + suffix over later chunks
    double S = s[i] + Boff[b];
    double c = 0.0;
    if (t < T_MAX) {
        float e = E[t];
        if (e != 0.0f) c = (double)e * log(S);
    }
    __syncthreads();
    s[i] = c;
    __syncthreads();
    for (int st = SCAN_BLK / 2; st > 0; st >>= 1) {
        if (i < st) s[i] += s[i + st];
        __syncthreads();
    }
    if (i == 0) atomic_add_f64(&acc[1], s[0]);
}

// ---------------------------------------------------------------- finalize
__global__ void cox_finalize(const double* __restrict__ acc,
                             float* __restrict__ out) {
    double loss = acc[1] - acc[0];            // sum E*logS - sum ev*x
    out[0] = (float)sqrt(loss / (double)N_ELEM);
}

// ---------------------------------------------------------------- launch
extern "C" void kernel_launch(void* const* d_in, const int* in_sizes, int n_in,
                              void* d_out, int out_size, void* d_ws, size_t ws_size,
                              hipStream_t stream) {
    const float* x  = (const float*)d_in[0];
    const float* tg = (const float*)d_in[1];
    int n  = in_sizes[0];
    int n4 = n / 4;

    char*   ws   = (char*)d_ws;
    float*  A    = (float*)(ws);
    float*  E    = (float*)(ws + 4000000);
    double* acc  = (double*)(ws + 8000000);
    double* Bsum = (double*)(ws + 8000016);
    double* Boff = (double*)(ws + 8008208);

    cox_zero<<<(ZERO_UINT4 + 255) / 256, 256, 0, stream>>>((v4f*)ws);
    cox_hist<<<(n4 + 255) / 256, 256, 0, stream>>>(
        (const v4f*)x, (const v4f*)tg, A, E, acc, n4);
    cox_block_sum<<<NB, 256, 0, stream>>>(A, Bsum);
    cox_scan_blocks<<<1, 1024, 0, stream>>>(Bsum, Boff);
    cox_suffix_logsum<<<NB, SCAN_BLK, 0, stream>>>(A, E, Boff, acc);
    cox_finalize<<<1, 1, 0, stream>>>(acc, (float*)d_out);
}